// WordAttention_68298569941186
// MI455X (gfx1250) — compile-verified
//
#include <hip/hip_runtime.h>
#include <hip/hip_bf16.h>

// Problem dims (from reference): A=4, B=16, T=1024, D=1024, UNITS=1024.
// Reference softmax is over a singleton axis -> l == 1.0 everywhere, so
//   out[b,a,d] = sum_t value[a,b,t,d]
// Pure bandwidth-bound column reduction (256 MB read @ 23.3 TB/s ~ 11 us).
// The reduction runs on the matrix unit: colsum = ones(16x4) x B(4x16) + C
// via V_WMMA_F32_16X16X4_F32 (fp32-exact; with A == 1 the result is
// invariant to the exact K placement inside the B fragment, so only the
// documented N = lane&15 mapping matters).

#define A_DIM 4
#define B_DIM 16
#define T_DIM 1024
#define D_DIM 1024

typedef float v8f __attribute__((ext_vector_type(8)));
typedef float v2f __attribute__((ext_vector_type(2)));

// ---------------------------------------------------------------------------
// Phase A: each block reduces a 128-row T-chunk of one (a,b) slab.
// grid = (A*B) * (T/128) = 64 * 8 = 512 blocks, 256 threads = 8 wave32.
// Wave w owns rows [t0 + 16w, t0 + 16w + 16). Each iteration handles TWO
// 16-column strips with independent accumulator chains (8 WMMAs, 16-load
// clauses) for ILP. Every row of each D tile equals the column sum, and the
// two half-waves hold identical copies, so the LDS publish is an
// unconditional same-value store (no EXEC manipulation).
// ---------------------------------------------------------------------------
__global__ __launch_bounds__(256) void colsum_wmma_kernel(
    const float* __restrict__ value, float* __restrict__ partial) {
  __shared__ float part8[8 * D_DIM];  // 32 KB: per-wave column sums

  const int tid  = threadIdx.x;
  const int w    = tid >> 5;   // wave id 0..7
  const int lane = tid & 31;

  const int bx = blockIdx.x;
  const int ab = bx >> 3;      // a*16 + b, 0..63
  const int tc = bx & 7;       // T-chunk 0..7 (128 rows each)

  // Base of this wave's 16 rows inside value[a][b][t][d]
  const float* base =
      value + ((size_t)ab * T_DIM + (size_t)tc * 128 + (size_t)w * 16) * D_DIM;

  const int n  = lane & 15;    // N (column within strip), matches C/D layout
  const int hl = lane >> 4;    // half-wave select for B K-placement

  const v2f ones = {1.0f, 1.0f};  // A-matrix 16x4 of 1.0 in every lane

  for (int d0 = 0; d0 < D_DIM; d0 += 32) {
    v8f acc0 = {0.f, 0.f, 0.f, 0.f, 0.f, 0.f, 0.f, 0.f};
    v8f acc1 = {0.f, 0.f, 0.f, 0.f, 0.f, 0.f, 0.f, 0.f};
    const float* col0 = base + d0 + n;
    const float* col1 = col0 + 16;
#pragma unroll
    for (int kg = 0; kg < 4; ++kg) {
      // B fragment (4x16): this lane's VGPR pair supplies rows
      // kg*4 + 2*hl + {0,1} at column n.
      const size_t r0 = (size_t)(kg * 4 + 2 * hl + 0) * D_DIM;
      const size_t r1 = (size_t)(kg * 4 + 2 * hl + 1) * D_DIM;
      v2f b0, b1;
      b0[0] = col0[r0];
      b0[1] = col0[r1];
      b1[0] = col1[r0];
      b1[1] = col1[r1];
      acc0 = __builtin_amdgcn_wmma_f32_16x16x4_f32(
          false, ones, false, b0, (short)0, acc0, false, false);
      acc1 = __builtin_amdgcn_wmma_f32_16x16x4_f32(
          false, ones, false, b1, (short)0, acc1, false, false);
    }
    // All D rows equal the column sum; both half-waves store the same value
    // to the same address (benign, deterministic).
    part8[w * D_DIM + d0 + n]      = acc0[0];
    part8[w * D_DIM + d0 + 16 + n] = acc1[0];
  }

  __syncthreads();

  // Combine the 8 per-wave partials -> 128-row partial sums for this block.
  for (int d = tid; d < D_DIM; d += 256) {
    float s = 0.f;
#pragma unroll
    for (int ww = 0; ww < 8; ++ww) s += part8[ww * D_DIM + d];
    partial[(size_t)bx * D_DIM + d] = s;
  }
}

// ---------------------------------------------------------------------------
// Phase B: sum the 8 T-chunk partials per (a,b) and write out[b][a][d].
// grid = 64 blocks (one per ab), 256 threads, float4 per thread.
// ---------------------------------------------------------------------------
__global__ __launch_bounds__(256) void combine_kernel(
    const float* __restrict__ partial, float* __restrict__ out) {
  const int ab = blockIdx.x;
  const int a  = ab >> 4;      // ab = a*16 + b
  const int b  = ab & 15;
  const int d  = threadIdx.x * 4;

  float s0 = 0.f, s1 = 0.f, s2 = 0.f, s3 = 0.f;
#pragma unroll
  for (int tc = 0; tc < 8; ++tc) {
    const float4 v =
        *(const float4*)(partial + ((size_t)(ab * 8 + tc)) * D_DIM + d);
    s0 += v.x; s1 += v.y; s2 += v.z; s3 += v.w;
  }
  float4 r; r.x = s0; r.y = s1; r.z = s2; r.w = s3;
  *(float4*)(out + ((size_t)(b * A_DIM + a)) * D_DIM + d) = r;
}

// ---------------------------------------------------------------------------
// Inputs (setup_inputs order): 0=query 1=value 2=W5_w 3=W5_b 4=W6_w 5=W6_b
// 6=V_w 7=V_b. Only `value` affects the output (softmax over a size-1 axis
// makes l == 1); V_b shifts scores uniformly and is absorbed by softmax too.
// ---------------------------------------------------------------------------
extern "C" void kernel_launch(void* const* d_in, const int* in_sizes, int n_in,
                              void* d_out, int out_size, void* d_ws,
                              size_t ws_size, hipStream_t stream) {
  const float* value = (const float*)d_in[1];
  float* out = (float*)d_out;
  float* partial = (float*)d_ws;  // 512 * 1024 fp32 = 2 MB scratch

  colsum_wmma_kernel<<<A_DIM * B_DIM * (T_DIM / 128), 256, 0, stream>>>(value,
                                                                        partial);
  combine_kernel<<<A_DIM * B_DIM, 256, 0, stream>>>(partial, out);
}